// HWDConv_20710332302087
// MI455X (gfx1250) — compile-verified
//
#include <hip/hip_runtime.h>
#include <hip/hip_bf16.h>

typedef float v2f __attribute__((ext_vector_type(2)));
typedef float v8f __attribute__((ext_vector_type(8)));
typedef int v4i_vs __attribute__((vector_size(16)));   // matches async-LDS builtin param

#define C_IN   256
#define H_IN   256
#define W_IN   256
#define HW     16384          // 128*128 output spatial
#define KDIM   1024           // 4*C channels after Haar concat
#define ODIM   512
#define GN_EPS 1e-5f

// workspace layout (float offsets)
#define Y_OFF    0                          // 1024*16384 = 16,777,216
#define PS_OFF   (Y_OFF + (size_t)KDIM*HW)  // 8192 partial sums
#define PQ_OFF   (PS_OFF + 8192)            // 8192 partial sumsq
#define A_OFF    (PQ_OFF + 8192)            // 1024
#define B_OFF    (A_OFF + 1024)             // 1024
#define W2_OFF   (B_OFF + 1024)             // 512*1024
#define BIAS_OFF (W2_OFF + (size_t)ODIM*KDIM) // 512
// total ~17.3M floats (~70 MB) of d_ws required

// CDNA5 async global->LDS staging (ASYNCcnt-tracked), with sync fallback.
#if defined(__AMDGCN__) && __has_builtin(__builtin_amdgcn_global_load_async_to_lds_b128) && __has_builtin(__builtin_amdgcn_s_wait_asynccnt)
#define ASYNC_STAGE 1
#else
#define ASYNC_STAGE 0
#endif

// ---------------------------------------------------------------------------
// Kernel 1: Haar bands + per-(band,channel,chunk) partial stats.
// ---------------------------------------------------------------------------
__global__ __launch_bounds__(256) void haar_kernel(const float* __restrict__ x,
                                                   float* __restrict__ y,
                                                   float* __restrict__ psum,
                                                   float* __restrict__ psq) {
  const int c = blockIdx.x;
  const int chunk = blockIdx.y;
  const int tid = threadIdx.x;
  const float* xc = x + (size_t)c * (H_IN * W_IN);

  float s[4] = {0.f, 0.f, 0.f, 0.f};
  float q[4] = {0.f, 0.f, 0.f, 0.f};

#pragma unroll
  for (int j = 0; j < 8; ++j) {
    int sidx = chunk * 2048 + j * 256 + tid;   // consecutive lanes -> consecutive w
    int h = sidx >> 7;
    int w = sidx & 127;
    const float2 r0 = *(const float2*)(xc + (size_t)(2 * h) * W_IN + 2 * w);
    const float2 r1 = *(const float2*)(xc + (size_t)(2 * h + 1) * W_IN + 2 * w);
    float x00 = r0.x, x01 = r0.y, x10 = r1.x, x11 = r1.y;
    float band[4];
    band[0] = 0.5f * ( x00 + x01 + x10 + x11);  // LL
    band[1] = 0.5f * (-x00 - x01 + x10 + x11);  // HL
    band[2] = 0.5f * (-x00 + x01 - x10 + x11);  // LH
    band[3] = 0.5f * ( x00 - x01 - x10 + x11);  // HH
#pragma unroll
    for (int b = 0; b < 4; ++b) {
      y[(size_t)(b * C_IN + c) * HW + sidx] = band[b];
      s[b] += band[b];
      q[b] += band[b] * band[b];
    }
  }

  __shared__ float redS[256];
  __shared__ float redQ[256];
#pragma unroll
  for (int b = 0; b < 4; ++b) {
    redS[tid] = s[b];
    redQ[tid] = q[b];
    __syncthreads();
    for (int st = 128; st > 0; st >>= 1) {
      if (tid < st) {
        redS[tid] += redS[tid + st];
        redQ[tid] += redQ[tid + st];
      }
      __syncthreads();
    }
    if (tid == 0) {
      psum[b * 2048 + c * 8 + chunk] = redS[0];
      psq [b * 2048 + c * 8 + chunk] = redQ[0];
    }
    __syncthreads();
  }
}

// ---------------------------------------------------------------------------
// Kernel 2: group stats -> per-channel affine (a, b).  grid = 4.
// ---------------------------------------------------------------------------
__global__ __launch_bounds__(256) void finalize_kernel(const float* __restrict__ psum,
                                                       const float* __restrict__ psq,
                                                       const float* __restrict__ gscale,
                                                       const float* __restrict__ gbias,
                                                       float* __restrict__ av,
                                                       float* __restrict__ bv) {
  int ch = blockIdx.x * 256 + threadIdx.x;   // 0..1023
  int g = ch >> 3;
  int base = (g >> 5) * 2048 + (g & 31) * 64;
  float S = 0.f, Q = 0.f;
  for (int i = 0; i < 64; ++i) {
    S += psum[base + i];
    Q += psq[base + i];
  }
  const float inv = 1.0f / 131072.0f;        // 8 ch * 128 * 128
  float mean = S * inv;
  float var = Q * inv - mean * mean;
  float rs = rsqrtf(var + GN_EPS);
  float a = gscale[ch] * rs;
  av[ch] = a;
  bv[ch] = gbias[ch] - mean * a;
}

// ---------------------------------------------------------------------------
// Kernel 3: fold GN affine into the projection.  grid = 512.
// ---------------------------------------------------------------------------
__global__ __launch_bounds__(256) void fold_kernel(const float* __restrict__ w,
                                                   const float* __restrict__ av,
                                                   const float* __restrict__ bv,
                                                   float* __restrict__ w2,
                                                   float* __restrict__ bias) {
  int o = blockIdx.x, tid = threadIdx.x;
  float acc = 0.f;
  for (int ch = tid; ch < KDIM; ch += 256) {
    float wv = w[(size_t)o * KDIM + ch];
    w2[(size_t)o * KDIM + ch] = wv * av[ch];
    acc += wv * bv[ch];
  }
  __shared__ float red[256];
  red[tid] = acc;
  __syncthreads();
  for (int st = 128; st > 0; st >>= 1) {
    if (tid < st) red[tid] += red[tid + st];
    __syncthreads();
  }
  if (tid == 0) bias[o] = red[0];
}

// ---------------------------------------------------------------------------
// Kernel 4: out = relu(w2 @ y + bias) via V_WMMA_F32_16X16X4_F32.
// Block tile 128(M) x 128(N), BK=16, double-buffered LDS with CDNA5
// GLOBAL_LOAD_ASYNC_TO_LDS_B128 staging (ASYNCcnt) when available.
// 8 wave32s in a 4x2 grid; each wave computes a 32x64 tile (2x4 accs).
// As[m][k] pad 20 (conflict-free b64 frag reads); Bs[k][n] pad 132
// (contiguous b128 async staging; frag reads merge to ds_load_2addr_b32).
// ---------------------------------------------------------------------------
#define BM 128
#define BN 128
#define BKC 16
#define LDA 20    // 16 + 4 pad: 16B-aligned rows, conflict-free b64 reads
#define LDB2 132  // 128 + 4 pad: 16B-aligned rows
#define NCH (KDIM / BKC)

__device__ __forceinline__ void stage_tiles(const float* __restrict__ w2,
                                            const float* __restrict__ y,
                                            float* __restrict__ AsBuf,
                                            float* __restrict__ BsBuf,
                                            int mBase, int nBase, int kk, int tid) {
  // A tile: 128 x 16 = 512 float4, 2 per thread (contiguous along K)
#pragma unroll
  for (int i = 0; i < 2; ++i) {
    int linear = tid + i * 256;
    int m = linear >> 2;               // 4 float4 per row
    int k4 = (linear & 3) << 2;
    const float* g = w2 + (size_t)(mBase + m) * KDIM + kk + k4;
    float* l = &AsBuf[m * LDA + k4];
#if ASYNC_STAGE
    __builtin_amdgcn_global_load_async_to_lds_b128((v4i_vs*)g, (v4i_vs*)l, 0, 0);
#else
    *(float4*)l = *(const float4*)g;
#endif
  }
  // B tile: 16 x 128 = 512 float4, 2 per thread (contiguous along N)
#pragma unroll
  for (int i = 0; i < 2; ++i) {
    int linear = tid + i * 256;
    int k = linear >> 5;               // 32 float4 per row
    int n4 = (linear & 31) << 2;
    const float* g = y + (size_t)(kk + k) * HW + nBase + n4;
    float* l = &BsBuf[k * LDB2 + n4];
#if ASYNC_STAGE
    __builtin_amdgcn_global_load_async_to_lds_b128((v4i_vs*)g, (v4i_vs*)l, 0, 0);
#else
    *(float4*)l = *(const float4*)g;
#endif
  }
}

__global__ __launch_bounds__(256) void gemm_kernel(const float* __restrict__ w2,
                                                   const float* __restrict__ y,
                                                   const float* __restrict__ bias,
                                                   float* __restrict__ out) {
  __shared__ float As[2][BM * LDA];
  __shared__ float Bs[2][BKC * LDB2];

  const int tid = threadIdx.x;
  const int lane = tid & 31;
  const int wave = tid >> 5;
  const int waveM = wave >> 1;   // 0..3 -> M offset waveM*32
  const int waveN = wave & 1;    // 0..1 -> N offset waveN*64
  const int nBase = blockIdx.x * BN;
  const int mBase = blockIdx.y * BM;

  v8f acc[2][4];
#pragma unroll
  for (int mi = 0; mi < 2; ++mi)
#pragma unroll
    for (int ni = 0; ni < 4; ++ni)
      acc[mi][ni] = (v8f){0.f, 0.f, 0.f, 0.f, 0.f, 0.f, 0.f, 0.f};

  const int mrow = waveM * 32 + (lane & 15);
  const int ncol = waveN * 64 + (lane & 15);
  const int khalf = (lane >> 4) << 1;      // lanes 16-31 hold K+2,K+3

  // prefetch chunk 0 into buffer 0
  stage_tiles(w2, y, As[0], Bs[0], mBase, nBase, 0, tid);

  for (int c = 0; c < NCH; ++c) {
    const int buf = c & 1;
    if (c + 1 < NCH) {
      stage_tiles(w2, y, As[buf ^ 1], Bs[buf ^ 1], mBase, nBase, (c + 1) * BKC, tid);
#if ASYNC_STAGE
      __builtin_amdgcn_s_wait_asynccnt(4);   // chunk c's 4 copies done; c+1 in flight
#endif
    } else {
#if ASYNC_STAGE
      __builtin_amdgcn_s_wait_asynccnt(0);
#endif
    }
    __syncthreads();

#pragma unroll
    for (int kr = 0; kr < BKC; kr += 4) {
      int kf = kr + khalf;
      v2f afrag[2], bfrag[4];
      afrag[0] = *(const v2f*)&As[buf][mrow * LDA + kf];
      afrag[1] = *(const v2f*)&As[buf][(mrow + 16) * LDA + kf];
#pragma unroll
      for (int ni = 0; ni < 4; ++ni) {
        int nc = ncol + 16 * ni;
        bfrag[ni] = (v2f){Bs[buf][kf * LDB2 + nc], Bs[buf][(kf + 1) * LDB2 + nc]};
      }
#pragma unroll
      for (int mi = 0; mi < 2; ++mi)
#pragma unroll
        for (int ni = 0; ni < 4; ++ni)
          acc[mi][ni] = __builtin_amdgcn_wmma_f32_16x16x4_f32(
              false, afrag[mi], false, bfrag[ni],
              (short)0, acc[mi][ni], false, false);
    }
    __syncthreads();   // all waves done with buf before it is overwritten
  }

  // --- epilogue: +bias, ReLU, store (C layout: VGPR r -> M=r / r+8) ---
  const int mhalf = (lane >> 4) << 3;
#pragma unroll
  for (int mi = 0; mi < 2; ++mi) {
#pragma unroll
    for (int ni = 0; ni < 4; ++ni) {
      int nGlob = nBase + waveN * 64 + ni * 16 + (lane & 15);
#pragma unroll
      for (int r = 0; r < 8; ++r) {
        int oGlob = mBase + waveM * 32 + mi * 16 + mhalf + r;
        float v = acc[mi][ni][r] + bias[oGlob];
        out[(size_t)oGlob * HW + nGlob] = fmaxf(v, 0.0f);
      }
    }
  }
}

// ---------------------------------------------------------------------------
extern "C" void kernel_launch(void* const* d_in, const int* in_sizes, int n_in,
                              void* d_out, int out_size, void* d_ws, size_t ws_size,
                              hipStream_t stream) {
  const float* x      = (const float*)d_in[0];   // 256*256*256
  const float* gscale = (const float*)d_in[1];   // 1024
  const float* gbias  = (const float*)d_in[2];   // 1024
  const float* w      = (const float*)d_in[3];   // 512*1024
  float* out = (float*)d_out;                    // 512*16384
  float* ws  = (float*)d_ws;                     // needs ~70 MB

  float* y    = ws + Y_OFF;
  float* psum = ws + PS_OFF;
  float* psq  = ws + PQ_OFF;
  float* av   = ws + A_OFF;
  float* bv   = ws + B_OFF;
  float* w2   = ws + W2_OFF;
  float* bias = ws + BIAS_OFF;

  haar_kernel<<<dim3(256, 8), 256, 0, stream>>>(x, y, psum, psq);
  finalize_kernel<<<dim3(4), 256, 0, stream>>>(psum, psq, gscale, gbias, av, bv);
  fold_kernel<<<dim3(512), 256, 0, stream>>>(w, av, bv, w2, bias);
  gemm_kernel<<<dim3(HW / BN, ODIM / BM), 256, 0, stream>>>(w2, y, bias, out);
}